// VectorQuantizer2_21225728377442
// MI455X (gfx1250) — compile-verified
//
#include <hip/hip_runtime.h>

typedef __attribute__((ext_vector_type(2))) float v2f;
typedef __attribute__((ext_vector_type(8))) float v8f;

// Problem constants (fixed by the reference)
#define BB   16
#define CC   32
#define HH   64
#define WW   64
#define VV   4096
#define HW   4096          // H*W
#define NPOS 65536         // B*H*W
#define NELEM 2097152      // B*C*H*W

// ---------------------------------------------------------------------------
// Kernel A: hn[v] = 0.5 * ||emb_w[v]||^2
// ---------------------------------------------------------------------------
__global__ void vq_halfnorm_kernel(const float* __restrict__ emb,
                                   float* __restrict__ hn) {
    int v = blockIdx.x * blockDim.x + threadIdx.x;   // 0..4095, exact grid
    const float4* r = reinterpret_cast<const float4*>(emb + v * CC);
    float s = 0.f;
#pragma unroll
    for (int i = 0; i < 8; ++i) {
        float4 e = r[i];
        s += e.x * e.x + e.y * e.y + e.z * e.z + e.w * e.w;
    }
    hn[v] = 0.5f * s;
}

// ---------------------------------------------------------------------------
// Kernel B: nearest-codebook search via fp32 WMMA, 2 row tiles per wave.
// One wave32 owns 32 positions (two 16-row A tiles). For each 16-code tile:
//   D[16x16] = Xtile[16x32] * EtileT[32x16] via 8 chained wmma_f32_16x16x4_f32
// B fragments are loaded once and reused for both A tiles (halves VMEM
// traffic per WMMA). score = x.e - 0.5||e||^2 ; argmax == argmin distance.
// C/D layout: lane l, vgpr v -> M = v + 8*(l>=16), N = l%16.
// A layout (16x4 f32): lane l -> M = l%16, vgpr j -> K = 2*(l/16) + j.
// ---------------------------------------------------------------------------
__global__ void vq_argmin_kernel(const float* __restrict__ f,
                                 const float* __restrict__ emb,
                                 const float* __restrict__ hn,
                                 int* __restrict__ idx_out) {
    const int lane  = threadIdx.x & 31;
    const int wave  = threadIdx.x >> 5;
    const int tile2 = blockIdx.x * 8 + wave;         // 0..2047 double tiles
    const int m     = lane & 15;                     // row (A) / col (B,D)
    const int hsel  = lane >> 4;                     // K-half selector

    // Two A row tiles: positions n0 = tile2*32 + m, n1 = n0 + 16
    const int n0 = tile2 * 32 + m;
    const int n1 = n0 + 16;
    const int b0 = n0 >> 12, r0 = n0 & (HW - 1);
    const int b1 = n1 >> 12, r1 = n1 & (HW - 1);
    const int fb0 = b0 * (CC * HW) + r0;             // + k*HW walks channels
    const int fb1 = b1 * (CC * HW) + r1;

    v2f a0[8], a1[8];
#pragma unroll
    for (int kc = 0; kc < 8; ++kc) {
        int k = kc * 4 + hsel * 2;
        v2f t0, t1;
        t0.x = f[fb0 + k * HW];
        t0.y = f[fb0 + (k + 1) * HW];
        t1.x = f[fb1 + k * HW];
        t1.y = f[fb1 + (k + 1) * HW];
        a0[kc] = t0;
        a1[kc] = t1;
    }

    float best0[8], best1[8];
    int   bidx0[8], bidx1[8];
#pragma unroll
    for (int v = 0; v < 8; ++v) {
        best0[v] = -3.4e38f; bidx0[v] = 0;
        best1[v] = -3.4e38f; bidx1[v] = 0;
    }

    for (int ct = 0; ct < 256; ++ct) {               // 256 code tiles of 16
        const int cbase = ct * 16;
        const float* er = emb + (cbase + m) * CC;    // this lane's code row
        v2f bf[8];
#pragma unroll
        for (int kc = 0; kc < 8; ++kc) {
            int k = kc * 4 + hsel * 2;
            bf[kc] = *reinterpret_cast<const v2f*>(er + k);   // 8B aligned
        }

        v8f acc0 = {0.f, 0.f, 0.f, 0.f, 0.f, 0.f, 0.f, 0.f};
        v8f acc1 = {0.f, 0.f, 0.f, 0.f, 0.f, 0.f, 0.f, 0.f};
#pragma unroll
        for (int kc = 0; kc < 8; ++kc) {
            acc0 = __builtin_amdgcn_wmma_f32_16x16x4_f32(
                false, a0[kc], false, bf[kc], (short)0, acc0, false, false);
            acc1 = __builtin_amdgcn_wmma_f32_16x16x4_f32(
                false, a1[kc], false, bf[kc], (short)0, acc1, false, false);
        }

        const float hnv = hn[cbase + m];             // col code = cbase + l%16
        const int   ci  = cbase + m;
#pragma unroll
        for (int v = 0; v < 8; ++v) {
            float s0 = acc0[v] - hnv;
            float s1 = acc1[v] - hnv;
            if (s0 > best0[v]) { best0[v] = s0; bidx0[v] = ci; }
            if (s1 > best1[v]) { best1[v] = s1; bidx1[v] = ci; }
        }
    }

    // Cross-lane reduction over the 16 columns (xor 8,4,2,1 stays in-half).
    // Tie-break: smaller index wins (matches jnp.argmin first-min).
#pragma unroll
    for (int v = 0; v < 8; ++v) {
        float bv0 = best0[v]; int bi0 = bidx0[v];
        float bv1 = best1[v]; int bi1 = bidx1[v];
#pragma unroll
        for (int off = 8; off > 0; off >>= 1) {
            float ov0 = __shfl_xor(bv0, off, 32);
            int   oi0 = __shfl_xor(bi0, off, 32);
            float ov1 = __shfl_xor(bv1, off, 32);
            int   oi1 = __shfl_xor(bi1, off, 32);
            if (ov0 > bv0 || (ov0 == bv0 && oi0 < bi0)) { bv0 = ov0; bi0 = oi0; }
            if (ov1 > bv1 || (ov1 == bv1 && oi1 < bi1)) { bv1 = ov1; bi1 = oi1; }
        }
        if ((lane & 15) == 0) {
            int row = v + (hsel << 3);               // M = v + 8*(l>=16)
            idx_out[tile2 * 32 + row]      = bi0;
            idx_out[tile2 * 32 + 16 + row] = bi1;
        }
    }
}

// ---------------------------------------------------------------------------
// Kernel C: gather h = emb[idx], Phi = 0.5*h + 0.5*(conv3x3(h)+bias),
// write f_hat, per-block partial sum of (f_hat - f)^2.
// Grid: (16 spatial tiles, 16 batches), 256 threads.
// ---------------------------------------------------------------------------
__global__ void vq_conv_kernel(const float* __restrict__ f,
                               const float* __restrict__ emb,
                               const float* __restrict__ cw,   // [32][32][3][3]
                               const float* __restrict__ cb,   // [32]
                               const int* __restrict__ idx,
                               float* __restrict__ out,
                               float* __restrict__ partials) {
    __shared__ float h_lds[CC * 18 * 18];    // 41472 B
    __shared__ float red[256];

    const int tid  = threadIdx.x;
    const int b    = blockIdx.y;
    const int ty   = blockIdx.x >> 2;
    const int tx   = blockIdx.x & 3;
    const int y0   = ty * 16;
    const int x0   = tx * 16;

    // Gather 18x18 halo tile, 32 channels each, zero-padded (SAME).
    for (int p = tid; p < 18 * 18; p += 256) {
        const int yy = p / 18, xx = p - yy * 18;
        const int gy = y0 + yy - 1, gx = x0 + xx - 1;
        if (gy >= 0 && gy < HH && gx >= 0 && gx < WW) {
            const int code = idx[(b * HH + gy) * WW + gx];
            const float4* er = reinterpret_cast<const float4*>(emb + code * CC);
#pragma unroll
            for (int c4 = 0; c4 < 8; ++c4) {
                float4 e = er[c4];
                h_lds[(c4 * 4 + 0) * 324 + p] = e.x;
                h_lds[(c4 * 4 + 1) * 324 + p] = e.y;
                h_lds[(c4 * 4 + 2) * 324 + p] = e.z;
                h_lds[(c4 * 4 + 3) * 324 + p] = e.w;
            }
        } else {
#pragma unroll
            for (int c = 0; c < CC; ++c) h_lds[c * 324 + p] = 0.f;
        }
    }
    __syncthreads();

    const int y = tid >> 4, x = tid & 15;    // each thread: one pixel, all co
    float sumsq = 0.f;

    for (int co = 0; co < CC; ++co) {        // co uniform -> scalar weights
        float acc = cb[co];
        const float* wrow = cw + co * (CC * 9);
#pragma unroll 4
        for (int ci = 0; ci < CC; ++ci) {
            const float* hc = &h_lds[ci * 324 + y * 18 + x];
            const float* wc = wrow + ci * 9;
            acc += wc[0] * hc[0]  + wc[1] * hc[1]  + wc[2] * hc[2]
                 + wc[3] * hc[18] + wc[4] * hc[19] + wc[5] * hc[20]
                 + wc[6] * hc[36] + wc[7] * hc[37] + wc[8] * hc[38];
        }
        const float hcen = h_lds[co * 324 + (y + 1) * 18 + (x + 1)];
        const float fh   = 0.5f * hcen + 0.5f * acc;   // RESI = 0.5
        const int gidx = ((b * CC + co) * HH + (y0 + y)) * WW + (x0 + x);
        out[gidx] = fh;
        const float d = fh - f[gidx];
        sumsq += d * d;
    }

    // Deterministic block reduction -> one partial per block.
    red[tid] = sumsq;
    __syncthreads();
    for (int s = 128; s > 0; s >>= 1) {
        if (tid < s) red[tid] += red[tid + s];
        __syncthreads();
    }
    if (tid == 0) partials[b * 16 + blockIdx.x] = red[0];
}

// ---------------------------------------------------------------------------
// Kernel D: loss = 1.25 * sum(partials) / NELEM   (BETA*mse + mse)
// ---------------------------------------------------------------------------
__global__ void vq_loss_kernel(const float* __restrict__ partials,
                               float* __restrict__ out) {
    __shared__ float red[256];
    const int tid = threadIdx.x;
    red[tid] = partials[tid];
    __syncthreads();
    for (int s = 128; s > 0; s >>= 1) {
        if (tid < s) red[tid] += red[tid + s];
        __syncthreads();
    }
    if (tid == 0) out[NELEM] = 1.25f * red[0] / (float)NELEM;
}

// ---------------------------------------------------------------------------
extern "C" void kernel_launch(void* const* d_in, const int* in_sizes, int n_in,
                              void* d_out, int out_size, void* d_ws, size_t ws_size,
                              hipStream_t stream) {
    const float* f      = (const float*)d_in[0];   // [16,32,64,64]
    const float* emb    = (const float*)d_in[1];   // [4096,32]
    const float* conv_w = (const float*)d_in[2];   // [32,32,3,3]
    const float* conv_b = (const float*)d_in[3];   // [32]
    float* out = (float*)d_out;                    // f_hat (2097152) + loss (1)

    float* hn       = (float*)d_ws;                          // 4096 f
    int*   idx      = (int*)((char*)d_ws + VV * sizeof(float));      // 65536 i
    float* partials = (float*)((char*)d_ws + VV * sizeof(float)
                                           + NPOS * sizeof(int));    // 256 f

    vq_halfnorm_kernel<<<VV / 256, 256, 0, stream>>>(emb, hn);
    vq_argmin_kernel<<<NPOS / 32 / 8, 256, 0, stream>>>(f, emb, hn, idx);
    vq_conv_kernel<<<dim3(16, BB), 256, 0, stream>>>(f, emb, conv_w, conv_b,
                                                     idx, out, partials);
    vq_loss_kernel<<<1, 256, 0, stream>>>(partials, out);
}